// MechanismGrabber_28887950033381
// MI455X (gfx1250) — compile-verified
//
#include <hip/hip_runtime.h>
#include <math.h>

#define DDIM 256
#define MMECH 64
#define NTOK 32768
#define TWOD 512

typedef float v2f __attribute__((ext_vector_type(2)));
typedef float v8f __attribute__((ext_vector_type(8)));

__device__ __forceinline__ v8f wmma_f32(v2f a, v2f b, v8f c) {
  return __builtin_amdgcn_wmma_f32_16x16x4_f32(false, a, false, b, (short)0, c,
                                               false, false);
}

// ---------------------------------------------------------------------------
// Kernel 1: selector + gate  ->  a[n, m] = softmax_m(W2·gelu(W1·ctx+b1)+b2) *
//                                          sigmoid(um·x+vm)
// block = 256 threads = 8 waves, 8 tokens per block (1 token per wave for the
// logit/softmax stage).
// ---------------------------------------------------------------------------
__global__ void __launch_bounds__(256) kSel(
    const float* __restrict__ x, const float* __restrict__ ctx,
    const float* __restrict__ um, const float* __restrict__ vm,
    const float* __restrict__ W1, const float* __restrict__ b1,
    const float* __restrict__ W2, const float* __restrict__ b2,
    float* __restrict__ aMat) {
  __shared__ float sctx[8 * DDIM];
  __shared__ float sx[8 * DDIM];
  __shared__ float sh[8 * TWOD];

  const int tid = threadIdx.x;
  const int tb = blockIdx.x * 8;

#pragma unroll
  for (int i = 0; i < 8; ++i) {
    int idx = tid + 256 * i;
    sctx[idx] = ctx[tb * DDIM + idx];
    sx[idx] = x[tb * DDIM + idx];
  }
  __syncthreads();

  // h = gelu(ctx @ W1^T + b1): 8 tokens x 512 outputs, 16 per thread
#pragma unroll
  for (int jj = 0; jj < 16; ++jj) {
    int o = tid + 256 * jj;       // 0..4095
    int tok = o >> 9;
    int j = o & (TWOD - 1);
    const float4* w4 = (const float4*)(W1 + j * DDIM);
    const float4* c4 = (const float4*)(sctx + tok * DDIM);
    float s = 0.f;
#pragma unroll 8
    for (int q = 0; q < DDIM / 4; ++q) {
      float4 a = w4[q];
      float4 b = c4[q];
      s += a.x * b.x + a.y * b.y + a.z * b.z + a.w * b.w;
    }
    s += b1[j];
    // exact GELU: 0.5*s*(1+erf(s/sqrt(2)))
    sh[tok * TWOD + j] = 0.5f * s * (1.0f + erff(s * 0.70710678118654752f));
  }
  __syncthreads();

  const int w = tid >> 5;
  const int lane = tid & 31;
  const int tok = w;

  float sl[2], gl[2];
#pragma unroll
  for (int t = 0; t < 2; ++t) {
    int m = lane + 32 * t;
    const float4* w2 = (const float4*)(W2 + m * TWOD);
    const float4* h4 = (const float4*)(sh + tok * TWOD);
    float s = 0.f;
#pragma unroll 8
    for (int q = 0; q < TWOD / 4; ++q) {
      float4 a = w2[q];
      float4 b = h4[q];
      s += a.x * b.x + a.y * b.y + a.z * b.z + a.w * b.w;
    }
    sl[t] = s + b2[m];
    const float4* u4 = (const float4*)(um + m * DDIM);
    const float4* x4 = (const float4*)(sx + tok * DDIM);
    float g = 0.f;
#pragma unroll 8
    for (int q = 0; q < DDIM / 4; ++q) {
      float4 a = u4[q];
      float4 b = x4[q];
      g += a.x * b.x + a.y * b.y + a.z * b.z + a.w * b.w;
    }
    gl[t] = g + vm[m];
  }
  // softmax over 64 mechanisms within one wave32 (2 values per lane)
  float mx = fmaxf(sl[0], sl[1]);
#pragma unroll
  for (int off = 16; off >= 1; off >>= 1) mx = fmaxf(mx, __shfl_xor(mx, off, 32));
  float e0 = expf(sl[0] - mx);
  float e1 = expf(sl[1] - mx);
  float ssum = e0 + e1;
#pragma unroll
  for (int off = 16; off >= 1; off >>= 1) ssum += __shfl_xor(ssum, off, 32);
  float inv = 1.0f / ssum;
  float g0 = 1.0f / (1.0f + expf(-gl[0]));
  float g1 = 1.0f / (1.0f + expf(-gl[1]));
  aMat[(tb + tok) * MMECH + lane] = e0 * inv * g0;
  aMat[(tb + tok) * MMECH + lane + 32] = e1 * inv * g1;
}

// ---------------------------------------------------------------------------
// Kernel 2: pack Wm / (bm+cm) / Wi into per-lane WMMA-B fragment order.
// Layout per 2 K-steps (sp): 32 lanes x float4 = 512B; lane l holds
// B[k=(l<16?0:2)+v, n=l&15] for s=2sp (v4 0,1) and s=2sp+1 (v4 2,3).
// ---------------------------------------------------------------------------
#define WPACK_N  (MMECH * DDIM * DDIM)           // 4,194,304
#define BCPACK_N (16 * 8 * 32 * 4)               // 16,384
#define WIPACK_N (16 * 64 * 32 * 4)              // 131,072
#define PACK_TOTAL (WPACK_N + BCPACK_N + WIPACK_N)

__global__ void __launch_bounds__(256) kPack(
    const float* __restrict__ Wm, const float* __restrict__ bm,
    const float* __restrict__ cm, const float* __restrict__ Wi,
    float* __restrict__ wpack, float* __restrict__ bcpack,
    float* __restrict__ wipack) {
  int tid = blockIdx.x * 256 + threadIdx.x;
  if (tid < WPACK_N) {
    int p = tid;
    int v4 = p & 3, l = (p >> 2) & 31, sp = (p >> 7) & 31;
    int m = (p >> 12) & 63, c = p >> 18;
    int s = 2 * sp + (v4 >> 1);
    int k = ((l < 16) ? 0 : 2) + (v4 & 1);
    int e = 4 * s + k;
    int d = 16 * c + (l & 15);
    wpack[p] = Wm[(m << 16) + (d << 8) + e];
  } else if (tid < WPACK_N + BCPACK_N) {
    int p = tid - WPACK_N;
    int v4 = p & 3, l = (p >> 2) & 31, sp = (p >> 7) & 7, c = p >> 10;
    int s = 2 * sp + (v4 >> 1);
    int k = ((l < 16) ? 0 : 2) + (v4 & 1);
    int mi = 4 * s + k;                    // K index == mechanism index
    int d = 16 * c + (l & 15);
    bcpack[p] = bm[mi * DDIM + d] + cm[mi * DDIM + d];
  } else if (tid < PACK_TOTAL) {
    int p = tid - WPACK_N - BCPACK_N;
    int v4 = p & 3, l = (p >> 2) & 31, sp = (p >> 7) & 63, c = p >> 13;
    int s = 2 * sp + (v4 >> 1);
    int k = ((l < 16) ? 0 : 2) + (v4 & 1);
    int e = 4 * s + k;                     // 0..511 over concat dim
    int d = 16 * c + (l & 15);
    wipack[p] = Wi[d * TWOD + e];
  }
}

// ---------------------------------------------------------------------------
// Kernel 3: selected[n,d] = sum_{m,e} (a[n,m]*x[n,e]) * Wm[m,d,e]
//                         + sum_m a[n,m]*(bm+cm)[m,d]
// One big K=16384 GEMM with on-the-fly A synthesis. Wave = 16 tokens x 256
// outs (16 f32 C tiles). Block = 4 waves = 64 tokens (64 KB dynamic LDS, safe
// under runtime caps; 2+ blocks co-resident per WGP for WGP$ reuse of wpack).
// x rows staged into LDS pre-swizzled into A-fragment order.
// ---------------------------------------------------------------------------
__global__ void __launch_bounds__(128) kHeavy(
    const float* __restrict__ x, const float* __restrict__ aMat,
    const float* __restrict__ wpack, const float* __restrict__ bcpack,
    float* __restrict__ selected) {
  extern __shared__ float lds[];
  const int tid = threadIdx.x;
  const int w = tid >> 5;
  const int lane = tid & 31;
  const int row0 = (blockIdx.x * 4 + w) * 16;
  float* xs = lds + w * 4096;
  const int rsel = lane & 15;
  const int o = (lane < 16) ? 0 : 2;

  // stage x rows into A-fragment order
  const float* xrow = x + (size_t)(row0 + rsel) * DDIM;
#pragma unroll 4
  for (int sp = 0; sp < 32; ++sp) {
    float2 lo = *(const float2*)(xrow + 8 * sp + o);
    float2 hi = *(const float2*)(xrow + 8 * sp + 4 + o);
    *(float4*)(xs + sp * 128 + lane * 4) = make_float4(lo.x, lo.y, hi.x, hi.y);
  }
  __syncthreads();

  v8f acc[16];
#pragma unroll
  for (int c = 0; c < 16; ++c) acc[c] = (v8f){};

  const float* arow = aMat + (size_t)(row0 + rsel) * MMECH;

  for (int m = 0; m < MMECH; ++m) {
    const float asc = arow[m];
    const float* wbase = wpack + (size_t)m * 4096 + lane * 4;
    for (int sp = 0; sp < 32; ++sp) {
      float4 xv = *(const float4*)(xs + sp * 128 + lane * 4);
      v2f A0 = {xv.x * asc, xv.y * asc};
      v2f A1 = {xv.z * asc, xv.w * asc};
      const float* wp = wbase + sp * 128;
#pragma unroll
      for (int c = 0; c < 16; ++c) {
        float4 bv = *(const float4*)(wp + (size_t)c * 262144);
        acc[c] = wmma_f32(A0, (v2f){bv.x, bv.y}, acc[c]);
        acc[c] = wmma_f32(A1, (v2f){bv.z, bv.w}, acc[c]);
      }
    }
  }

  // bias GEMM: K = 64 over mechanisms, A = a[n,m], B = bm+cm
#pragma unroll
  for (int sp = 0; sp < 8; ++sp) {
    float2 lo = *(const float2*)(arow + 8 * sp + o);
    float2 hi = *(const float2*)(arow + 8 * sp + 4 + o);
    v2f A0 = {lo.x, lo.y};
    v2f A1 = {hi.x, hi.y};
#pragma unroll
    for (int c = 0; c < 16; ++c) {
      float4 bv = *(const float4*)(bcpack + (c * 8 + sp) * 128 + lane * 4);
      acc[c] = wmma_f32(A0, (v2f){bv.x, bv.y}, acc[c]);
      acc[c] = wmma_f32(A1, (v2f){bv.z, bv.w}, acc[c]);
    }
  }

  const int rowhi = (lane >> 4) * 8;
#pragma unroll
  for (int c = 0; c < 16; ++c) {
#pragma unroll
    for (int r = 0; r < 8; ++r) {
      selected[(size_t)(row0 + r + rowhi) * DDIM + 16 * c + rsel] = acc[c][r];
    }
  }
}

// ---------------------------------------------------------------------------
// Kernel 4: out = concat(x, selected) @ Wi^T + bi   (K = 512, two LDS phases)
// ---------------------------------------------------------------------------
__global__ void __launch_bounds__(128) kInt(
    const float* __restrict__ x, const float* __restrict__ selected,
    const float* __restrict__ wipack, const float* __restrict__ bi,
    float* __restrict__ out) {
  extern __shared__ float lds[];
  const int tid = threadIdx.x;
  const int w = tid >> 5;
  const int lane = tid & 31;
  const int row0 = (blockIdx.x * 4 + w) * 16;
  float* xs = lds + w * 4096;
  const int rsel = lane & 15;
  const int o = (lane < 16) ? 0 : 2;

  v8f acc[16];
#pragma unroll
  for (int c = 0; c < 16; ++c) acc[c] = (v8f){};

  // ---- phase 1: e in [0,256) from x ----
  const float* xrow = x + (size_t)(row0 + rsel) * DDIM;
#pragma unroll 4
  for (int sp = 0; sp < 32; ++sp) {
    float2 lo = *(const float2*)(xrow + 8 * sp + o);
    float2 hi = *(const float2*)(xrow + 8 * sp + 4 + o);
    *(float4*)(xs + sp * 128 + lane * 4) = make_float4(lo.x, lo.y, hi.x, hi.y);
  }
  __syncthreads();
  for (int sp = 0; sp < 32; ++sp) {
    float4 xv = *(const float4*)(xs + sp * 128 + lane * 4);
    v2f A0 = {xv.x, xv.y};
    v2f A1 = {xv.z, xv.w};
#pragma unroll
    for (int c = 0; c < 16; ++c) {
      float4 bv = *(const float4*)(wipack + (c * 64 + sp) * 128 + lane * 4);
      acc[c] = wmma_f32(A0, (v2f){bv.x, bv.y}, acc[c]);
      acc[c] = wmma_f32(A1, (v2f){bv.z, bv.w}, acc[c]);
    }
  }
  __syncthreads();

  // ---- phase 2: e in [256,512) from selected ----
  const float* srow = selected + (size_t)(row0 + rsel) * DDIM;
#pragma unroll 4
  for (int sp = 0; sp < 32; ++sp) {
    float2 lo = *(const float2*)(srow + 8 * sp + o);
    float2 hi = *(const float2*)(srow + 8 * sp + 4 + o);
    *(float4*)(xs + sp * 128 + lane * 4) = make_float4(lo.x, lo.y, hi.x, hi.y);
  }
  __syncthreads();
  for (int sp = 0; sp < 32; ++sp) {
    float4 xv = *(const float4*)(xs + sp * 128 + lane * 4);
    v2f A0 = {xv.x, xv.y};
    v2f A1 = {xv.z, xv.w};
#pragma unroll
    for (int c = 0; c < 16; ++c) {
      float4 bv = *(const float4*)(wipack + (c * 64 + 32 + sp) * 128 + lane * 4);
      acc[c] = wmma_f32(A0, (v2f){bv.x, bv.y}, acc[c]);
      acc[c] = wmma_f32(A1, (v2f){bv.z, bv.w}, acc[c]);
    }
  }

  const int rowhi = (lane >> 4) * 8;
#pragma unroll
  for (int c = 0; c < 16; ++c) {
    float b = bi[16 * c + rsel];
#pragma unroll
    for (int r = 0; r < 8; ++r) {
      out[(size_t)(row0 + r + rowhi) * DDIM + 16 * c + rsel] = acc[c][r] + b;
    }
  }
}

// ---------------------------------------------------------------------------
extern "C" void kernel_launch(void* const* d_in, const int* in_sizes, int n_in,
                              void* d_out, int out_size, void* d_ws,
                              size_t ws_size, hipStream_t stream) {
  const float* x   = (const float*)d_in[0];
  const float* ctx = (const float*)d_in[1];
  const float* Wm  = (const float*)d_in[2];
  const float* bm  = (const float*)d_in[3];
  const float* cm  = (const float*)d_in[4];
  const float* um  = (const float*)d_in[5];
  const float* vm  = (const float*)d_in[6];
  const float* W1  = (const float*)d_in[7];
  const float* b1  = (const float*)d_in[8];
  const float* W2  = (const float*)d_in[9];
  const float* b2  = (const float*)d_in[10];
  const float* Wi  = (const float*)d_in[11];
  const float* bi  = (const float*)d_in[12];

  float* ws = (float*)d_ws;
  float* aMat     = ws;                        // N*M        = 2,097,152
  float* selected = ws + 2097152;              // N*D        = 8,388,608
  float* wpack    = ws + 10485760;             // WPACK_N    = 4,194,304
  float* bcpack   = ws + 14680064;             // BCPACK_N   = 16,384
  float* wipack   = ws + 14696448;             // WIPACK_N   = 131,072

  kSel<<<NTOK / 8, 256, 0, stream>>>(x, ctx, um, vm, W1, b1, W2, b2, aMat);
  kPack<<<PACK_TOTAL / 256, 256, 0, stream>>>(Wm, bm, cm, Wi, wpack, bcpack,
                                              wipack);
  kHeavy<<<NTOK / 64, 128, 65536, stream>>>(x, aMat, wpack, bcpack, selected);
  kInt<<<NTOK / 64, 128, 65536, stream>>>(x, selected, wipack, bi,
                                          (float*)d_out);
}